// DeformationGrid_65180423684269
// MI455X (gfx1250) — compile-verified
//
#include <hip/hip_runtime.h>
#include <stdint.h>

// ---------------------------------------------------------------------------
// Trilinear grid interpolation, MI455X (gfx1250).
// Memory-bound gather kernel: grid (25MB) lives in L2 (192MB); coords/out
// stream from HBM. Coords are staged LDS-side via the Tensor Data Mover
// (tensor_load_to_lds, TENSORcnt) with double buffering so the vector-memory
// pipe is dedicated to the 8 corner gathers per point.
// ---------------------------------------------------------------------------

typedef unsigned int v4u __attribute__((ext_vector_type(4)));
typedef int          v8i __attribute__((ext_vector_type(8)));
typedef int          v4i __attribute__((ext_vector_type(4)));

#define BLOCK_THREADS   256
#define TILE_PTS        2048
#define TILE_WORDS      (TILE_PTS * 3)              // 6144 dwords = 24 KB
#define PTS_PER_THREAD  (TILE_PTS / BLOCK_THREADS)  // 8

struct F3 { float x, y, z; };

// Build a Tensor DMA Descriptor (D#) for a 1-D dword copy of `TILE_WORDS`
// words starting at `gptr` into LDS byte offset `lds_off`, and issue
// TENSOR_LOAD_TO_LDS.  tensor_dim0 = remaining_words makes any tail reads
// OOB -> zero-filled instead of faulting.
__device__ __forceinline__ void tdm_load_tile(uint32_t lds_off,
                                              const float* gptr,
                                              uint32_t remaining_words) {
  uint64_t ga = (uint64_t)(uintptr_t)gptr;

  // D# group 0: [1:0]=count=1, [63:32]=lds_addr, [120:64]=global_addr,
  //             [127:126]=type=2 ("image")
  v4u g0;
  g0[0] = 1u;
  g0[1] = lds_off;
  g0[2] = (uint32_t)ga;
  g0[3] = ((uint32_t)(ga >> 32) & 0x01FFFFFFu) | (2u << 30);

  // D# group 1 (bit positions per ISA 8.4):
  //  w0: workgroup_mask=0, data_size=2 (4B) at [17:16]
  //  tensor_dim0 [79:48]  -> w1[31:16] = lo16, w2[15:0] = hi16
  //  tensor_dim1 [111:80] -> w2[31:16] = 1
  //  tile_dim0  [127:112] -> w3[31:16] = TILE_WORDS
  //  tile_dim1/2 unused (0); tensor_dim0_stride [207:160] -> w5 (sane value)
  uint32_t d0 = remaining_words;
  v8i g1;
  g1[0] = (int)(2u << 16);
  g1[1] = (int)(d0 << 16);
  g1[2] = (int)((d0 >> 16) | (1u << 16));
  g1[3] = (int)((uint32_t)TILE_WORDS << 16);
  g1[4] = 0;
  g1[5] = (int)d0;
  g1[6] = 0;
  g1[7] = 0;

  // D# group 2: tensor_dim2=1, tensor_dim3=1, stride2=0, tile_dim3=0(unused)
  v4i g2; g2[0] = 1; g2[1] = 1; g2[2] = 0; g2[3] = 0;
  // D# group 3: stride3=0, tensor_dim4=1 ([79:48] -> w1[31:16]), tile_dim4=0
  v4i g3; g3[0] = 0; g3[1] = (int)(1u << 16); g3[2] = 0; g3[3] = 0;

  // 6-arg toolchain variant: extra SGPR group beyond the 4 architectural D#
  // groups (VADDR4 is unused/NULL per ISA) — zero-filled.
  v8i g4; g4[0] = 0; g4[1] = 0; g4[2] = 0; g4[3] = 0;
          g4[4] = 0; g4[5] = 0; g4[6] = 0; g4[7] = 0;

  __builtin_amdgcn_tensor_load_to_lds(g0, g1, g2, g3, g4, /*cpol*/0);
}

__device__ __forceinline__ void interp_point(const float* __restrict__ theta,
                                             float ux, float uy, float uz,
                                             float* __restrict__ outp) {
  const float hi = 1.0f - 1e-7f;
  ux = fminf(fmaxf(ux, 0.0f), hi);
  uy = fminf(fmaxf(uy, 0.0f), hi);
  uz = fminf(fmaxf(uz, 0.0f), hi);

  float fx = ux * 127.0f, fy = uy * 127.0f, fz = uz * 127.0f;
  int ix = (int)fx, iy = (int)fy, iz = (int)fz;   // floor (non-negative)
  float wx1 = fx - (float)ix, wy1 = fy - (float)iy, wz1 = fz - (float)iz;
  float wx0 = 1.0f - wx1,     wy0 = 1.0f - wy1,    wz0 = 1.0f - wz1;

  const int DX = 128 * 128 * 3, DY = 128 * 3, DZ = 3;
  const float* g = theta + ((((ix << 7) + iy) << 7) + iz) * 3;

  // 8 corner gathers, 12B each (global_load_b96) — L2-resident grid.
  F3 c000 = *(const F3*)(g);
  F3 c100 = *(const F3*)(g + DX);
  F3 c010 = *(const F3*)(g + DY);
  F3 c110 = *(const F3*)(g + DX + DY);
  F3 c001 = *(const F3*)(g + DZ);
  F3 c101 = *(const F3*)(g + DX + DZ);
  F3 c011 = *(const F3*)(g + DY + DZ);
  F3 c111 = *(const F3*)(g + DX + DY + DZ);

  float w000 = wx0 * wy0 * wz0, w100 = wx1 * wy0 * wz0;
  float w010 = wx0 * wy1 * wz0, w110 = wx1 * wy1 * wz0;
  float w001 = wx0 * wy0 * wz1, w101 = wx1 * wy0 * wz1;
  float w011 = wx0 * wy1 * wz1, w111 = wx1 * wy1 * wz1;

  float rx = c000.x * w000, ry = c000.y * w000, rz = c000.z * w000;
  rx = fmaf(c100.x, w100, rx); ry = fmaf(c100.y, w100, ry); rz = fmaf(c100.z, w100, rz);
  rx = fmaf(c010.x, w010, rx); ry = fmaf(c010.y, w010, ry); rz = fmaf(c010.z, w010, rz);
  rx = fmaf(c110.x, w110, rx); ry = fmaf(c110.y, w110, ry); rz = fmaf(c110.z, w110, rz);
  rx = fmaf(c001.x, w001, rx); ry = fmaf(c001.y, w001, ry); rz = fmaf(c001.z, w001, rz);
  rx = fmaf(c101.x, w101, rx); ry = fmaf(c101.y, w101, ry); rz = fmaf(c101.z, w101, rz);
  rx = fmaf(c011.x, w011, rx); ry = fmaf(c011.y, w011, ry); rz = fmaf(c011.z, w011, rz);
  rx = fmaf(c111.x, w111, rx); ry = fmaf(c111.y, w111, ry); rz = fmaf(c111.z, w111, rz);

  F3 r; r.x = rx; r.y = ry; r.z = rz;
  *(F3*)outp = r;
}

__global__ __launch_bounds__(BLOCK_THREADS)
void trilerp_tdm_kernel(const float* __restrict__ coords,
                        const float* __restrict__ theta,
                        float* __restrict__ out,
                        int npts) {
  __shared__ float sbuf[2][TILE_WORDS];   // 48 KB double buffer

  // LDS byte offsets of the two staging buffers (generic ptr low 32 bits are
  // the LDS aperture offset).
  const uint32_t lds0 = (uint32_t)(uintptr_t)&sbuf[0][0];
  const uint32_t lds1 = (uint32_t)(uintptr_t)&sbuf[1][0];

  const int tid = threadIdx.x;
  const bool wave0 = (tid >> 5) == 0;     // one TDM issue per block
  const uint32_t totalWords = (uint32_t)npts * 3u;
  const int nTiles = (npts + TILE_PTS - 1) / TILE_PTS;

  int t0 = blockIdx.x;
  if (t0 >= nTiles) return;

  if (wave0) {
    uint32_t startW = (uint32_t)t0 * (uint32_t)TILE_WORDS;
    tdm_load_tile(lds0, coords + startW, totalWords - startW);
  }

  int curBuf = 0;
  for (int t = t0; t < nTiles; t += gridDim.x) {
    int tn = t + gridDim.x;
    if (wave0) {
      if (tn < nTiles) {
        // Prefetch next tile into the other buffer, then wait for current.
        uint32_t startW = (uint32_t)tn * (uint32_t)TILE_WORDS;
        tdm_load_tile(curBuf ? lds0 : lds1, coords + startW, totalWords - startW);
        __builtin_amdgcn_s_wait_tensorcnt(1);  // TDM completes in order
      } else {
        __builtin_amdgcn_s_wait_tensorcnt(0);
      }
    }
    __syncthreads();                           // publish sbuf[curBuf]

    const float* s = &sbuf[curBuf][0];
    const int baseP = t * TILE_PTS;
#pragma unroll
    for (int k = 0; k < PTS_PER_THREAD; ++k) {
      int lp = tid + k * BLOCK_THREADS;
      int p = baseP + lp;
      if (p < npts) {
        float ux = s[3 * lp + 0];
        float uy = s[3 * lp + 1];
        float uz = s[3 * lp + 2];
        interp_point(theta, ux, uy, uz, out + 3 * (size_t)p);
      }
    }
    __syncthreads();                           // readers done before buf reuse
    curBuf ^= 1;
  }
}

extern "C" void kernel_launch(void* const* d_in, const int* in_sizes, int n_in,
                              void* d_out, int out_size, void* d_ws, size_t ws_size,
                              hipStream_t stream) {
  const float* coords = (const float*)d_in[0];   // [N,3] f32
  const float* theta  = (const float*)d_in[1];   // [128,128,128,3] f32
  float* out = (float*)d_out;                    // [N,3] f32

  int npts = in_sizes[0] / 3;
  int nTiles = (npts + TILE_PTS - 1) / TILE_PTS;
  int blocks = nTiles < 2048 ? nTiles : 2048;    // 2 tiles/block at N=2^23

  trilerp_tdm_kernel<<<blocks, BLOCK_THREADS, 0, stream>>>(coords, theta, out, npts);
}